// VanilaGCN_6038724018302
// MI455X (gfx1250) — compile-verified
//
#include <hip/hip_runtime.h>
#include <hip/hip_bf16.h>
#include <math.h>

typedef float v2f __attribute__((ext_vector_type(2)));
typedef float v8f __attribute__((ext_vector_type(8)));

#define HDIM 128

// ---------------- small utility kernels ----------------

__global__ void k_fill(float* __restrict__ p, float v, int n) {
    int t = blockIdx.x * blockDim.x + threadIdx.x;
    if (t < n) p[t] = v;
}

// sum of squares of edge weights -> *out (must be pre-zeroed)
__global__ void k_sumsq(const float* __restrict__ w, int E, float* __restrict__ out) {
    __shared__ float red[256];
    int t = blockIdx.x * blockDim.x + threadIdx.x;
    float v = 0.f;
    if (t < E) { float x = w[t]; v = x * x; }
    red[threadIdx.x] = v;
    __syncthreads();
    for (int s = 128; s > 0; s >>= 1) {
        if ((int)threadIdx.x < s) red[threadIdx.x] += red[threadIdx.x + s];
        __syncthreads();
    }
    if (threadIdx.x == 0) atomicAdd(out, red[0]);
}

// deg[col[e]] += ew[e] / max(||ew||, 1e-12)   (deg pre-filled with 1.0 = self loop)
__global__ void k_deg(const float* __restrict__ ew, const int* __restrict__ col,
                      float* __restrict__ deg, const float* __restrict__ sumsq, int E) {
    int t = blockIdx.x * blockDim.x + threadIdx.x;
    if (t >= E) return;
    float inv = 1.f / fmaxf(sqrtf(*sumsq), 1e-12f);
    atomicAdd(&deg[col[t]], ew[t] * inv);
}

// deg -> dis = rsqrt(deg) in place (deg >= 1 always, matches reference's guarded rsqrt)
__global__ void k_rsqrt(float* __restrict__ deg, int n) {
    int t = blockIdx.x * blockDim.x + threadIdx.x;
    if (t < n) deg[t] = rsqrtf(deg[t]);
}

// norm[e] = dis[row] * (ew/||ew||) * dis[col]
__global__ void k_norm(const float* __restrict__ ew, const int* __restrict__ row,
                       const int* __restrict__ col, const float* __restrict__ dis,
                       const float* __restrict__ sumsq, float* __restrict__ nv, int E) {
    int t = blockIdx.x * blockDim.x + threadIdx.x;
    if (t >= E) return;
    float inv = 1.f / fmaxf(sqrtf(*sumsq), 1e-12f);
    nv[t] = dis[row[t]] * (ew[t] * inv) * dis[col[t]];
}

// ---------------- WMMA fp32 GEMM: C[nrows x 128] = op(A)[nrows x 128] @ B[128 x 128] ----------------
// RELU_IN / BIAS / RELU_OUT are compile-time so the inner loop is branch-free
// (no v_cndmask around the relu; EXEC stays all-1s for WMMA).
// One wave per 16x16 output tile; 8 waves per block cover the 8 column tiles of one row tile.
template <bool RELU_IN, bool BIAS, bool RELU_OUT>
__global__ void k_gemm(const float* __restrict__ A, const float* __restrict__ B,
                       float* __restrict__ C, int nrows, const float* __restrict__ bias) {
    int lane = threadIdx.x & 31;
    int wv   = threadIdx.x >> 5;        // 0..7 -> column tile
    int lm   = lane & 15;
    int hi   = lane >> 4;               // 0 or 1
    int mtile = blockIdx.x;
    int col  = wv * 16 + lm;

    int r = mtile * 16 + lm;
    if (r >= nrows) r = nrows - 1;      // clamp loads; stores are guarded below
    const float* pa = A + (size_t)r * HDIM + 2 * hi;

    v8f acc = {};
#pragma unroll
    for (int k = 0; k < HDIM; k += 4) {
        __builtin_prefetch(pa + k + 16, 0, 3);
        v2f a = *(const v2f*)(pa + k);
        if (RELU_IN) { a.x = fmaxf(a.x, 0.f); a.y = fmaxf(a.y, 0.f); }
        v2f b;
        b.x = B[(size_t)(k + 2 * hi)     * HDIM + col];
        b.y = B[(size_t)(k + 2 * hi + 1) * HDIM + col];
        // D = A(16x4 f32) * B(4x16 f32) + C  -> v_wmma_f32_16x16x4_f32
        acc = __builtin_amdgcn_wmma_f32_16x16x4_f32(false, a, false, b,
                                                    (short)0, acc, false, false);
    }

    int rowbase = mtile * 16 + hi * 8;  // C layout: VGPR j holds M=j (lanes 0-15), M=j+8 (lanes 16-31)
    float bc = BIAS ? bias[col] : 0.f;
#pragma unroll
    for (int j = 0; j < 8; ++j) {
        int ro = rowbase + j;
        if (ro < nrows) {
            float v = acc[j];
            if (BIAS) v += bc;
            if (RELU_OUT) v = fmaxf(v, 0.f);
            C[(size_t)ro * HDIM + col] = v;
        }
    }
}

// agg[i][f] = bias[f] + dis[i]^2 * m[i][f]   (bias + self-loop message fused)
__global__ void k_initagg(const float* __restrict__ m, const float* __restrict__ dis,
                          const float* __restrict__ bias, float* __restrict__ agg, int total) {
    int t = blockIdx.x * blockDim.x + threadIdx.x;
    if (t >= total) return;
    int i = t >> 7, f = t & 127;
    float d = dis[i];
    agg[t] = bias[f] + d * d * m[t];
}

// edge scatter: agg[col[e]] += norm[e] * m[row[e]]; 32 lanes/edge, float4 per lane
__global__ void k_scatter(const float* __restrict__ m, const int* __restrict__ row,
                          const int* __restrict__ col, const float* __restrict__ nv,
                          float* __restrict__ agg, int E) {
    long long gid = (long long)blockIdx.x * blockDim.x + threadIdx.x;
    int e = (int)(gid >> 5);
    int lane = (int)(gid & 31);
    if (e >= E) return;
    int rs = row[e], cs = col[e];
    float w = nv[e];
    const float4 v = *(const float4*)(m + (size_t)rs * HDIM + lane * 4);
    float* dst = agg + (size_t)cs * HDIM + lane * 4;
    atomicAdd(dst + 0, w * v.x);
    atomicAdd(dst + 1, w * v.y);
    atomicAdd(dst + 2, w * v.z);
    atomicAdd(dst + 3, w * v.w);
}

// pooled[batch[i]][f] += relu(h[i][f])
__global__ void k_pool(const float* __restrict__ h, const int* __restrict__ batch,
                       float* __restrict__ gp, int total) {
    int t = blockIdx.x * blockDim.x + threadIdx.x;
    if (t >= total) return;
    int i = t >> 7, f = t & 127;
    atomicAdd(&gp[(size_t)batch[i] * HDIM + f], fmaxf(h[t], 0.f));
}

// final 128->10 linear + log_softmax; one block (32 threads) per graph
__global__ void k_head(const float* __restrict__ g2, const float* __restrict__ lw2,
                       const float* __restrict__ lb2, float* __restrict__ out, int C) {
    int g = blockIdx.x;
    __shared__ float lg[16];
    __shared__ float stats[2];
    int c = threadIdx.x;
    if (c < C) {
        float s = lb2[c];
        for (int k = 0; k < HDIM; ++k) s += g2[(size_t)g * HDIM + k] * lw2[k * C + c];
        lg[c] = s;
    }
    __syncthreads();
    if (c == 0) {
        float mx = lg[0];
        for (int i = 1; i < C; ++i) mx = fmaxf(mx, lg[i]);
        float se = 0.f;
        for (int i = 0; i < C; ++i) se += expf(lg[i] - mx);
        stats[0] = mx; stats[1] = logf(se);
    }
    __syncthreads();
    if (c < C) out[(size_t)g * C + c] = lg[c] - stats[0] - stats[1];
}

// ---------------- launcher ----------------

extern "C" void kernel_launch(void* const* d_in, const int* in_sizes, int n_in,
                              void* d_out, int out_size, void* d_ws, size_t ws_size,
                              hipStream_t stream) {
    const float* x   = (const float*)d_in[0];
    const int*   ei  = (const int*)d_in[1];   // [2, E]
    const float* ew  = (const float*)d_in[2];
    const int*   bat = (const int*)d_in[3];
    const float* W[5]  = {(const float*)d_in[4], (const float*)d_in[6], (const float*)d_in[8],
                          (const float*)d_in[10], (const float*)d_in[12]};
    const float* Bv[5] = {(const float*)d_in[5], (const float*)d_in[7], (const float*)d_in[9],
                          (const float*)d_in[11], (const float*)d_in[13]};
    const float* lw1 = (const float*)d_in[14];
    const float* lb1 = (const float*)d_in[15];
    const float* lw2 = (const float*)d_in[16];
    const float* lb2 = (const float*)d_in[17];
    float* out = (float*)d_out;

    const int E = in_sizes[2];
    const int N = in_sizes[0] / HDIM;
    const int C = 10;
    const int G = out_size / C;

    const int* row = ei;
    const int* col = ei + E;

    // workspace layout (256B aligned chunks)
    char* ws = (char*)d_ws;
    size_t off = 0;
    auto carve = [&](size_t bytes) { char* p = ws + off; off += (bytes + 255) & ~(size_t)255; return p; };
    float* sumsq = (float*)carve(sizeof(float));
    float* dis   = (float*)carve((size_t)N * sizeof(float));      // deg, then rsqrt'd in place
    float* nv    = (float*)carve((size_t)E * sizeof(float));
    float* mbuf  = (float*)carve((size_t)N * HDIM * sizeof(float));
    float* hA    = (float*)carve((size_t)N * HDIM * sizeof(float));
    float* hB    = (float*)carve((size_t)N * HDIM * sizeof(float));
    float* gp    = (float*)carve((size_t)G * HDIM * sizeof(float));
    float* g2    = (float*)carve((size_t)G * HDIM * sizeof(float));
    (void)ws_size; (void)n_in;

    const int BT = 256;
    const int totNH = N * HDIM;

    // --- gcn_norm precompute ---
    k_fill<<<1, 64, 0, stream>>>(sumsq, 0.f, 1);
    k_sumsq<<<(E + BT - 1) / BT, BT, 0, stream>>>(ew, E, sumsq);
    k_fill<<<(N + BT - 1) / BT, BT, 0, stream>>>(dis, 1.0f, N);   // self-loop weight
    k_deg<<<(E + BT - 1) / BT, BT, 0, stream>>>(ew, col, dis, sumsq, E);
    k_rsqrt<<<(N + BT - 1) / BT, BT, 0, stream>>>(dis, N);
    k_norm<<<(E + BT - 1) / BT, BT, 0, stream>>>(ew, row, col, dis, sumsq, nv, E);

    // --- 5 GCN layers (ReLU fused into next GEMM's A-load) ---
    const int gemmGrid = (N + 15) / 16;
    const int scatGrid = (int)(((long long)E * 32 + BT - 1) / BT);
    const float* hin = x;
    float* outs[5] = {hB, hA, hB, hA, hB};
    for (int l = 0; l < 5; ++l) {
        float* agg = outs[l];
        if (l == 0)
            k_gemm<false, false, false><<<gemmGrid, BT, 0, stream>>>(hin, W[l], mbuf, N, nullptr);
        else
            k_gemm<true,  false, false><<<gemmGrid, BT, 0, stream>>>(hin, W[l], mbuf, N, nullptr);
        k_initagg<<<(totNH + BT - 1) / BT, BT, 0, stream>>>(mbuf, dis, Bv[l], agg, totNH);
        k_scatter<<<scatGrid, BT, 0, stream>>>(mbuf, row, col, nv, agg, E);
        hin = agg;
    }

    // --- pool (with fused final ReLU) + MLP head ---
    k_fill<<<(G * HDIM + BT - 1) / BT, BT, 0, stream>>>(gp, 0.f, G * HDIM);
    k_pool<<<(totNH + BT - 1) / BT, BT, 0, stream>>>(hin, bat, gp, totNH);
    // relu(gp @ lw1 + lb1) on the matrix pipe as well
    k_gemm<false, true, true><<<(G + 15) / 16, BT, 0, stream>>>(gp, lw1, g2, G, lb1);
    k_head<<<G, 32, 0, stream>>>(g2, lw2, lb2, out, C);
}